// NaiveBias2d_36876589204103
// MI455X (gfx1250) — compile-verified
//
#include <hip/hip_runtime.h>
#include <hip/hip_bf16.h>

// Problem constants (from reference): n=8, L=1026, h=16, d=64, s=32
#define SS 32
#define LL 1026
#define HH 16
#define DD 64
#define DH 32   // d-columns per block (DD split across 2 blocks)

typedef float v2f __attribute__((ext_vector_type(2)));
typedef float v8f __attribute__((ext_vector_type(8)));

// One workgroup per (batch n, head h, d-half). 256 threads = 8 waves (wave32).
// Single pass over v (NT loads, each element read once grid-wide):
//   U[a,:] = sum_b v[n,1+a*32+b,h,:]   (each rowgroup owns 4 'a' rows -> direct)
//   T[b,:] = sum_a v[n,1+a*32+b,h,:]   (8-way LDS partials, deterministic reduce)
// Then P = bias @ U, Q = bias @ T via v_wmma_f32_16x16x4_f32 (fp32 exact),
// and pbv[n,1+c*32+dd,h,:] = P[c,:] + Q[dd,:] streamed out with NT stores.
__global__ __launch_bounds__(256) void pbv_kernel(const float* __restrict__ v,
                                                  const float* __restrict__ offset,
                                                  const float* __restrict__ w,
                                                  float* __restrict__ out) {
  __shared__ float wm[SS];
  __shared__ float biasL[SS][SS + 1];       // +1 pad for column reads
  __shared__ float Ul[SS][DH];
  __shared__ float Tl[SS][DH];
  __shared__ float Pl[SS][DH];
  __shared__ float Ql[SS][DH];
  __shared__ float Tpart[8][SS][DH];        // per-rowgroup T partials

  const int tid = threadIdx.x;
  const int bx  = blockIdx.x;               // n*HH*2 blocks
  const int nb  = bx >> 5;                  // batch index
  const int hh  = (bx >> 1) & 15;           // head index
  const int dh  = bx & 1;                   // d-half (0/1)

  // wm[k] = exp(w[h,k] - offset[h])
  if (tid < SS) wm[tid] = __expf(w[hh * SS + tid] - offset[hh]);
  __syncthreads();
  // bias[a][c] = wm[|a-c|]  (symmetric Toeplitz)
  for (int idx = tid; idx < SS * SS; idx += 256) {
    int a = idx >> 5, c = idx & 31;
    biasL[a][c] = wm[a > c ? a - c : c - a];
  }

  const int lcol = tid & 31;                // local d-column 0..31
  const int rgrp = tid >> 5;                // rowgroup 0..7 (== wave id)
  const float* vbase =
      v + ((size_t)nb * LL * HH + hh) * DD + dh * DH + lcol;  // + l*HH*DD

  // ---- Single pass over v: U direct, T via per-group partials ----
  float treg[SS];
  #pragma unroll
  for (int b = 0; b < SS; ++b) treg[b] = 0.f;

  #pragma unroll
  for (int ai = 0; ai < 4; ++ai) {
    const int a = rgrp * 4 + ai;            // this group's exclusive 'a' rows
    const float* vp = vbase + (size_t)(1 + a * SS) * HH * DD;
    float usum = 0.f;
    #pragma unroll
    for (int b = 0; b < SS; ++b) {
      const float x = __builtin_nontemporal_load(vp + (size_t)b * HH * DD);
      usum    += x;
      treg[b] += x;
    }
    Ul[a][lcol] = usum;
  }
  #pragma unroll
  for (int b = 0; b < SS; ++b) Tpart[rgrp][b][lcol] = treg[b];
  __syncthreads();

  // Deterministic 8-way reduce of T partials
  for (int idx = tid; idx < SS * DH; idx += 256) {
    const int b = idx >> 5, col = idx & 31;
    float s = 0.f;
    #pragma unroll
    for (int g = 0; g < 8; ++g) s += Tpart[g][b][col];
    Tl[b][col] = s;
  }
  __syncthreads();

  // ---- WMMA phase: 8 tiles (P: 2x2, Q: 2x2 of 16x16), one per wave ----
  const int lane  = tid & 31;
  const int mrow  = lane & 15;              // M (and N) position within tile
  const int khalf = (lane >> 4) << 1;       // lane-group K offset {0,2}
  const int isQ   = rgrp >> 2;              // waves 4..7 compute Q
  const int t2    = rgrp & 3;
  const int mi    = t2 >> 1;                // M tile (0..1)
  const int ni    = t2 & 1;                 // N tile (0..1)
  const float (*Sm)[DH] = isQ ? Tl : Ul;
  float (*Dm)[DH]       = isQ ? Ql : Pl;
  const int m    = mi * 16 + mrow;          // output row (c or dd)
  const int ncol = ni * 16 + mrow;          // B column
  v8f acc = {0.f, 0.f, 0.f, 0.f, 0.f, 0.f, 0.f, 0.f};
  #pragma unroll
  for (int k = 0; k < 8; ++k) {             // K = 32 in steps of 4
    const int k0 = k * 4 + khalf;
    v2f afrag, bfrag;
    afrag.x = biasL[m][k0];                 // A[M][K] (bias symmetric)
    afrag.y = biasL[m][k0 + 1];
    bfrag.x = Sm[k0][ncol];                 // B[K][N]
    bfrag.y = Sm[k0 + 1][ncol];
    acc = __builtin_amdgcn_wmma_f32_16x16x4_f32(
        /*neg_a=*/false, afrag, /*neg_b=*/false, bfrag,
        /*c_mod=*/(short)0, acc, /*reuse_a=*/false, /*reuse_b=*/false);
  }
  // D layout: lane L holds N=L%16; VGPR r holds M = r + 8*(L/16)
  const int rowoff = mi * 16 + ((lane >> 4) << 3);
  #pragma unroll
  for (int ri = 0; ri < 8; ++ri) Dm[rowoff + ri][ni * 16 + mrow] = acc[ri];
  __syncthreads();

  // ---- Stream pbv (NT stores; written once, never re-read) ----
  float* obase = out + ((size_t)nb * LL * HH + hh) * DD + dh * DH + lcol;
  for (int jr = rgrp; jr < SS * SS; jr += 8) {
    const int c = jr >> 5, dd = jr & 31;
    __builtin_nontemporal_store(Pl[c][lcol] + Ql[dd][lcol],
                                obase + (size_t)(1 + jr) * HH * DD);
  }
  if (rgrp == 0) {                          // zero-padded border rows
    __builtin_nontemporal_store(0.f, obase);
    __builtin_nontemporal_store(0.f, obase + (size_t)(LL - 1) * HH * DD);
  }
}

// z_pb[0, n, l, h] = s * (rowsum[h,a] + rowsum[h,b]), l = 1 + a*32 + b; borders 0.
__global__ __launch_bounds__(256) void zpb_kernel(const float* __restrict__ offset,
                                                  const float* __restrict__ w,
                                                  float* __restrict__ outz,
                                                  int total) {
  __shared__ float rowsum[HH][SS];
  const int tid = threadIdx.x;
  for (int idx = tid; idx < HH * SS; idx += 256) {
    const int hh = idx >> 5, a = idx & 31;
    float rs = 0.f;
    for (int c = 0; c < SS; ++c) {
      const int dlt = a > c ? a - c : c - a;
      rs += __expf(w[hh * SS + dlt] - offset[hh]);
    }
    rowsum[hh][a] = rs;
  }
  __syncthreads();
  const int gid = blockIdx.x * 256 + tid;
  if (gid < total) {
    const int hh = gid % HH;
    const int l  = (gid / HH) % LL;
    float val = 0.f;
    if (l > 0 && l < LL - 1) {
      const int lb = l - 1;
      val = (float)SS * (rowsum[hh][lb >> 5] + rowsum[hh][lb & 31]);
    }
    __builtin_nontemporal_store(val, outz + gid);
  }
}

extern "C" void kernel_launch(void* const* d_in, const int* in_sizes, int n_in,
                              void* d_out, int out_size, void* d_ws, size_t ws_size,
                              hipStream_t stream) {
  const float* v      = (const float*)d_in[0];  // (n, L, h, d) fp32
  const float* offset = (const float*)d_in[1];  // (1, h)       fp32
  const float* w      = (const float*)d_in[2];  // (1, h, s)    fp32
  float* out = (float*)d_out;                   // pbv then z_pb, fp32

  const int n = in_sizes[0] / (LL * HH * DD);   // = 8

  // pbv: one block per (n, h, d-half)
  pbv_kernel<<<n * HH * 2, 256, 0, stream>>>(v, offset, w, out);

  // z_pb broadcast: (1, n, L, h)
  const int totalz = n * LL * HH;
  float* outz = out + (size_t)n * LL * HH * DD;
  zpb_kernel<<<(totalz + 255) / 256, 256, 0, stream>>>(offset, w, outz, totalz);
}